// binning_CRVD_25864293056737
// MI455X (gfx1250) — compile-verified
//
#include <hip/hip_runtime.h>

// ---------------------------------------------------------------------------
// binning_CRVD forward, MI455X (gfx1250)
//
// Pure-bandwidth op (AI ~0.4 FLOP/B): floor = 335 MB / 23.3 TB/s ~= 14 us.
// Tensor Data Mover streams 4-row x 256-float tiles (the 4 tap channels of
// one output channel, row stride 16*HW) into a per-wave 3-deep LDS ring
// (2 prefetches outstanding per wave -> ~2x in-flight DMA vs double buffer);
// VALU does 4 FMAs/output; 128-bit non-temporal stores.
// WMMA deliberately NOT used: the 16x64 weight matrix has 1 nonzero per
// column (block-diagonal), so matrix ops add 16x wasted FLOPs + layout
// shuffles with zero bandwidth benefit.
// ---------------------------------------------------------------------------

typedef unsigned int v4u __attribute__((ext_vector_type(4)));
typedef int          v8i __attribute__((ext_vector_type(8)));
typedef int          v4i __attribute__((ext_vector_type(4)));
typedef float        v4f __attribute__((ext_vector_type(4)));

#define HW_PIX   65536                 // 256*256
#define SPLITS   4                     // segments per (b,c) plane
#define SEG      (HW_PIX / SPLITS)     // 16384 elements per block
#define NWAVES   8                     // 256 threads / wave32
#define WSEG     (SEG / NWAVES)        // 2048 elements per wave
#define TILE     256                   // elements per TDM row tile
#define NIT      (WSEG / TILE)         // 8 iterations per wave
#define BUFS     3                     // LDS ring depth (2 prefetches in flight)

// Issue one TDM descriptor: load a 2D tile (4 rows x TILE floats) from
// global (row stride = 16*HW_PIX elements) into LDS at lds_byte_addr.
// Bitfield packing per CDNA5 ISA ch.8 (D# group0/group1).
__device__ __forceinline__ void tdm_stage(const float* gptr, unsigned lds_byte_addr) {
  unsigned long long ga = (unsigned long long)gptr;

  v4u g0;
  g0[0] = 1u;                                   // count=1, is_restore=0, gather=0
  g0[1] = lds_byte_addr;                        // lds_addr[31:0]
  g0[2] = (unsigned)ga;                         // global_addr[31:0]
  g0[3] = (unsigned)((ga >> 32) & 0x01FFFFFFull)// global_addr[56:32]
          | 0x80000000u;                        // type=2 ("image")

  v8i g1;
  g1[0] = 0x00020000;          // workgroup_mask=0, data_size=2 (4B), no flags
  g1[1] = 0;                   // atomic_barrier=0, tensor_dim0[15:0]=0
  g1[2] = 0x00044000;          // tensor_dim0[31:16]=0x4000 (2^30), tensor_dim1[15:0]=4
  g1[3] = (TILE << 16);        // tensor_dim1[31:16]=0, tile_dim0=TILE
  g1[4] = 4;                   // tile_dim1=4, tile_dim2=0 (2D)
  g1[5] = 16 * HW_PIX;         // tensor_dim0_stride[31:0] = 1048576 elements
  g1[6] = 0;                   // stride hi, tensor_dim1_stride lo
  g1[7] = 0;                   // tensor_dim1_stride hi

  v4i gz = {0, 0, 0, 0};       // groups 2/3 unused (<=2D tensor)
#if __clang_major__ >= 23
  v8i z8 = {0, 0, 0, 0, 0, 0, 0, 0};
  __builtin_amdgcn_tensor_load_to_lds(g0, g1, gz, gz, z8, 0);
#else
  __builtin_amdgcn_tensor_load_to_lds(g0, g1, gz, gz, 0);
#endif
}

__global__ __launch_bounds__(256) void binning_crvd_kernel(
    const float* __restrict__ x,
    const float* __restrict__ t_gb, const float* __restrict__ t_b,
    const float* __restrict__ t_r,  const float* __restrict__ t_gr,
    float* __restrict__ out)
{
  // Per-wave 3-deep staging ring: [wave][buf][tap row][TILE] = 96 KB.
  __shared__ __align__(16) float lds[NWAVES][BUFS][4][TILE];

  const int blk   = (int)blockIdx.x;
  const int plane = blk / SPLITS;        // 0..255 : (b*16 + c)
  const int s     = blk % SPLITS;
  const int b     = plane >> 4;
  const int c     = plane & 15;
  const int g     = c >> 2;              // group: 0=gb 1=b 2=r 3=gr

  // Branchless uniform tap select (s_cselect chain, no branch ladder).
  const float* taps = t_gb;
  taps = (g == 1) ? t_b  : taps;
  taps = (g == 2) ? t_r  : taps;
  taps = (g == 3) ? t_gr : taps;
  const float w0 = taps[0], w1 = taps[1], w2 = taps[2], w3 = taps[3];

  const int wv   = __builtin_amdgcn_readfirstlane((int)(threadIdx.x >> 5));
  const int lane = (int)(threadIdx.x & 31u);

  const long long seg = (long long)s * SEG + (long long)wv * WSEG;
  const float* xin = x   + ((long long)(b * 64 + c) * HW_PIX + seg); // tap j at +j*16*HW
  float*       op  = out + ((long long)(b * 16 + c) * HW_PIX + seg);

  // LDS byte offset of this wave's buffer 0 (flat LDS addr low 32 bits).
  const unsigned lds0 =
      (unsigned)__builtin_amdgcn_readfirstlane(
          (int)(unsigned)(unsigned long long)&lds[wv][0][0][0]);
  const unsigned bufBytes = 4u * TILE * 4u;   // 4 KB per ring slot

  // Prologue: fill two ring slots.
  tdm_stage(xin, lds0);
  tdm_stage(xin + TILE, lds0 + bufBytes);

#pragma unroll
  for (int it = 0; it < NIT; ++it) {
    const int cur = it % BUFS;
    if (it + 2 < NIT) {
      // Keep two prefetches in flight; TDM ops complete in order per wave,
      // so tensorcnt<=2 means the op for the current slot has landed.
      tdm_stage(xin + (long long)(it + 2) * TILE,
                lds0 + (unsigned)((it + 2) % BUFS) * bufBytes);
      __builtin_amdgcn_s_wait_tensorcnt(2);
    } else if (it + 1 < NIT) {
      __builtin_amdgcn_s_wait_tensorcnt(1);
    } else {
      __builtin_amdgcn_s_wait_tensorcnt(0);
    }
    // TDM writes to LDS are invisible to the compiler: fence it.
    asm volatile("" ::: "memory");

    const v4f* r0 = (const v4f*)&lds[wv][cur][0][0];
    const v4f* r1 = (const v4f*)&lds[wv][cur][1][0];
    const v4f* r2 = (const v4f*)&lds[wv][cur][2][0];
    const v4f* r3 = (const v4f*)&lds[wv][cur][3][0];
    v4f* ov = (v4f*)(op + (long long)it * TILE);

#pragma unroll
    for (int u = 0; u < TILE / 128; ++u) {     // 2 float4 per lane
      const int kv = lane + u * 32;
      v4f a = r0[kv];                          // ds_load_b128 x4
      v4f e = r1[kv];
      v4f f = r2[kv];
      v4f h = r3[kv];
      v4f o = a * w0 + e * w1 + f * w2 + h * w3;   // contracts to v_fma_f32
      __builtin_nontemporal_store(o, &ov[kv]);     // global_store_b128 NT
    }
    asm volatile("" ::: "memory");
  }
}

extern "C" void kernel_launch(void* const* d_in, const int* in_sizes, int n_in,
                              void* d_out, int out_size, void* d_ws, size_t ws_size,
                              hipStream_t stream) {
  (void)in_sizes; (void)n_in; (void)out_size; (void)d_ws; (void)ws_size;
  const float* x    = (const float*)d_in[0];
  const float* bgb  = (const float*)d_in[1];
  const float* bb   = (const float*)d_in[2];
  const float* br   = (const float*)d_in[3];
  const float* bgr  = (const float*)d_in[4];
  float* out = (float*)d_out;

  dim3 grid(256 * SPLITS);   // 16 batches * 16 out-channels * SPLITS
  dim3 block(256);           // 8 wave32 waves
  binning_crvd_kernel<<<grid, block, 0, stream>>>(x, bgb, bb, br, bgr, out);
}